// GATNetV3_7670811591307
// MI455X (gfx1250) — compile-verified
//
#include <hip/hip_runtime.h>
#include <hip/hip_bf16.h>
#include <cstddef>

// ---------------------------------------------------------------------------
// CDNA5 (gfx1250) GAT network: WMMA f32 GEMMs fed by async global->LDS DMA,
// plus edge softmax/scatter kernels.
// ---------------------------------------------------------------------------

typedef float v2f __attribute__((ext_vector_type(2)));
typedef float v8f __attribute__((ext_vector_type(8)));

typedef __attribute__((address_space(1))) int* as1_i32p;
typedef __attribute__((address_space(3))) int* as3_i32p;

#define TM 64
#define TN 64
#define KC 32
#define LDA  34   // padded LDS stride for A tile  [row][k]   (KC + 2, even)
#define LDBT 34   // padded LDS stride for B tile  [col][k]   (KC + 2, even)

// ---- async global->LDS copy (gfx1250 GLOBAL_LOAD_ASYNC_TO_LDS_B32) --------
// Builtin signature (from clang diagnostic): (AS1 int* gsrc, AS3 int* lds,
// imm offset, imm cpol). Casts go through integers: always legal, strips
// const; AS3 value is the low 32 bits of the flat address per the aperture
// rule.
#if __has_builtin(__builtin_amdgcn_global_load_async_to_lds_b32)
#define ASYNC_CP_B32(gptr, ldsptr)                                            \
    __builtin_amdgcn_global_load_async_to_lds_b32(                            \
        (as1_i32p)(size_t)(gptr),                                             \
        (as3_i32p)(unsigned)(size_t)(ldsptr), 0, 0)
#else
#define ASYNC_CP_B32(gptr, ldsptr)                                            \
    asm volatile("global_load_async_to_lds_b32 %0, %1, off"                   \
                 :: "v"((unsigned)(size_t)(ldsptr)),                          \
                    "v"((const void*)(gptr))                                  \
                 : "memory")
#endif

#if __has_builtin(__builtin_amdgcn_s_wait_asynccnt)
#define WAIT_ASYNC() __builtin_amdgcn_s_wait_asynccnt(0)
#else
#define WAIT_ASYNC() asm volatile("s_wait_asynccnt 0" ::: "memory")
#endif

// ---------------------------------------------------------------------------
// GEMM: C[M,N] = A[M,K] @ B[K,N] (+bias, optional ReLU).  M % 64 == 0.
// Block = 128 threads = 4 waves. Block tile 64x64, wave tile 16x64 with four
// independent v_wmma_f32_16x16x4_f32 accumulator chains. Tiles staged into
// LDS via async DMA; B stored transposed so both fragments are ds_load_b64.
// ---------------------------------------------------------------------------
__global__ __launch_bounds__(128)
void gemm_wmma_f32(const float* __restrict__ A, const float* __restrict__ B,
                   const float* __restrict__ bias, float* __restrict__ C,
                   int M, int N, int K, int relu)
{
    __shared__ float As[TM * LDA];    // A tile, row-major [row][k]
    __shared__ float Bt[TN * LDBT];   // B tile, transposed [col][k]

    const int tid  = threadIdx.x;
    const int wave = tid >> 5;
    const int lane = tid & 31;
    const int m0 = blockIdx.x * TM;
    const int n0 = blockIdx.y * TN;

    const int rowm = wave * 16 + (lane & 15);   // A row (within block tile)
    const int koff = (lane >> 4) * 2;           // lanes 0-15 -> K0/K1, 16-31 -> K2/K3
    const int coln = lane & 15;                 // B/C column within 16-wide subtile

    v8f acc[4] = {};

    for (int k0 = 0; k0 < K; k0 += KC) {
        // Prefetch next B chunk (global_prefetch_b8 on gfx1250).
        if (k0 + KC < K) {
            int pr = (tid >> 6) + k0 + KC;
            int pc = (tid & 63) + n0;
            if (pr < K && pc < N)
                __builtin_prefetch(&B[(size_t)pr * N + pc], 0, 1);
        }
        // Stage A tile: TM x KC (async DMA; zero-fill pad via ds_store)
        #pragma unroll
        for (int i = tid; i < TM * KC; i += 128) {
            int r = i >> 5, c = i & 31;
            int gm = m0 + r, gk = k0 + c;
            float* ldst = &As[r * LDA + c];
            if (gm < M && gk < K) ASYNC_CP_B32(&A[(size_t)gm * K + gk], ldst);
            else                  *ldst = 0.f;
        }
        // Stage B tile transposed: Bt[col][k] (global reads coalesced over N)
        #pragma unroll
        for (int i = tid; i < KC * TN; i += 128) {
            int r = i >> 6, c = i & 63;
            int gk = k0 + r, gn = n0 + c;
            float* ldst = &Bt[c * LDBT + r];
            if (gk < K && gn < N) ASYNC_CP_B32(&B[(size_t)gk * N + gn], ldst);
            else                  *ldst = 0.f;
        }
        WAIT_ASYNC();
        __syncthreads();

        #pragma unroll
        for (int kk = 0; kk < KC; kk += 4) {
            const float* ap = &As[rowm * LDA + kk + koff];
            v2f a; a.x = ap[0]; a.y = ap[1];
            #pragma unroll
            for (int j = 0; j < 4; ++j) {
                const float* bp = &Bt[(j * 16 + coln) * LDBT + kk + koff];
                v2f b; b.x = bp[0]; b.y = bp[1];
                acc[j] = __builtin_amdgcn_wmma_f32_16x16x4_f32(
                    false, a, false, b, (short)0, acc[j], false, false);
            }
        }
        __syncthreads();
    }

    // Epilogue. C/D layout: VGPR r, lanes 0-15 -> M=r, lanes 16-31 -> M=r+8.
    const int crow0 = m0 + wave * 16 + (lane >> 4) * 8;
    #pragma unroll
    for (int j = 0; j < 4; ++j) {
        int gn = n0 + j * 16 + coln;
        if (gn < N) {
            float bv = bias ? bias[gn] : 0.f;
            #pragma unroll
            for (int r = 0; r < 8; ++r) {
                float v = acc[j][r] + bv;
                if (relu) v = fmaxf(v, 0.f);
                C[(size_t)(crow0 + r) * N + gn] = v;
            }
        }
    }
}

// ---------------------------------------------------------------------------
// Attention scores: als[n,h] = sum_c h[n,h,c]*a_src[h,c]; ald likewise. H=2.
// One wave32 per node.
// ---------------------------------------------------------------------------
__global__ __launch_bounds__(128)
void gat_scores(const float* __restrict__ h, const float* __restrict__ a_src,
                const float* __restrict__ a_dst, float* __restrict__ als,
                float* __restrict__ ald, int N, int C)
{
    int n    = blockIdx.x * 4 + (threadIdx.x >> 5);
    int lane = threadIdx.x & 31;
    if (n >= N) return;
    const float* hn = h + (size_t)n * (2 * C);
    #pragma unroll
    for (int hh = 0; hh < 2; ++hh) {
        float ss = 0.f, sd = 0.f;
        for (int c = lane; c < C; c += 32) {
            float v = hn[hh * C + c];
            ss += v * a_src[hh * C + c];
            sd += v * a_dst[hh * C + c];
        }
        #pragma unroll
        for (int off = 16; off > 0; off >>= 1) {
            ss += __shfl_down(ss, off, 32);
            sd += __shfl_down(sd, off, 32);
        }
        if (lane == 0) { als[n * 2 + hh] = ss; ald[n * 2 + hh] = sd; }
    }
}

// ---------------------------------------------------------------------------
// Per-layer state reset: m = -inf-ish, den = 0, agg = 0.
// ---------------------------------------------------------------------------
__global__ void gat_reset(float* __restrict__ mm, float* __restrict__ den,
                          float* __restrict__ agg, int NH, size_t NHC)
{
    size_t i = (size_t)blockIdx.x * 256 + threadIdx.x;
    if (i < (size_t)NH) { mm[i] = -3.0e38f; den[i] = 0.f; }
    if (i < NHC) agg[i] = 0.f;
}

// Monotonic-bit-pattern float atomic max (compiles to int/uint atomics).
__device__ __forceinline__ void atomicMaxFloat(float* addr, float v)
{
    if (!(v < 0.f)) atomicMax((int*)addr, __float_as_int(v));
    else            atomicMin((unsigned int*)addr, (unsigned int)__float_as_int(v));
}

// ---------------------------------------------------------------------------
// Edge pass 1: score = leaky_relu(als[src]+ald[dst]); segment max over dst.
// Edges e < E_in come from edge_index; e >= E_in are self-loops.
// ---------------------------------------------------------------------------
__global__ void edge_max_k(const int* __restrict__ ei, const float* __restrict__ als,
                           const float* __restrict__ ald, float* __restrict__ mm,
                           float* __restrict__ esc, int E_in, int E_tot)
{
    int e = blockIdx.x * 256 + threadIdx.x;
    if (e >= E_tot) return;
    int s, d;
    if (e < E_in) { s = ei[e]; d = ei[E_in + e]; }
    else          { s = e - E_in; d = s; }
    #pragma unroll
    for (int hh = 0; hh < 2; ++hh) {
        float sc = als[s * 2 + hh] + ald[d * 2 + hh];
        sc = (sc > 0.f) ? sc : 0.2f * sc;   // LeakyReLU(0.2)
        esc[(size_t)e * 2 + hh] = sc;
        atomicMaxFloat(&mm[d * 2 + hh], sc);
    }
}

// ---------------------------------------------------------------------------
// Edge pass 2: ex = exp(score - m[dst]); segment sum over dst; keep ex.
// ---------------------------------------------------------------------------
__global__ void edge_expsum_k(const int* __restrict__ ei, const float* __restrict__ mm,
                              float* __restrict__ esc, float* __restrict__ den,
                              int E_in, int E_tot)
{
    int e = blockIdx.x * 256 + threadIdx.x;
    if (e >= E_tot) return;
    int d = (e < E_in) ? ei[E_in + e] : (e - E_in);
    #pragma unroll
    for (int hh = 0; hh < 2; ++hh) {
        float ex = __expf(esc[(size_t)e * 2 + hh] - mm[d * 2 + hh]);
        esc[(size_t)e * 2 + hh] = ex;
        atomicAdd(&den[d * 2 + hh], ex);
    }
}

// ---------------------------------------------------------------------------
// Edge pass 3: agg[dst] += alpha * h[src].  One wave32 per edge, lanes over
// the 2*C feature channels (global_atomic_add_f32).
// ---------------------------------------------------------------------------
__global__ __launch_bounds__(128)
void edge_scatter_k(const int* __restrict__ ei, const float* __restrict__ esc,
                    const float* __restrict__ den, const float* __restrict__ h,
                    float* __restrict__ agg, int E_in, int E_tot, int C)
{
    int e    = blockIdx.x * 4 + (threadIdx.x >> 5);
    int lane = threadIdx.x & 31;
    if (e >= E_tot) return;
    int s, d;
    if (e < E_in) { s = ei[e]; d = ei[E_in + e]; }
    else          { s = e - E_in; d = s; }
    const int HC = 2 * C;
    float a0 = esc[(size_t)e * 2 + 0] / den[d * 2 + 0];
    float a1 = esc[(size_t)e * 2 + 1] / den[d * 2 + 1];
    const float* hs = h   + (size_t)s * HC;
    float*       od = agg + (size_t)d * HC;
    for (int f = lane; f < HC; f += 32) {
        float al = (f < C) ? a0 : a1;
        atomicAdd(&od[f], al * hs[f]);
    }
}

// ---------------------------------------------------------------------------
// x = relu(x + bias[col])
// ---------------------------------------------------------------------------
__global__ void bias_relu_k(float* __restrict__ x, const float* __restrict__ b,
                            size_t total, int HC)
{
    size_t i = (size_t)blockIdx.x * 256 + threadIdx.x;
    if (i < total) {
        float v = x[i] + b[(int)(i % (size_t)HC)];
        x[i] = fmaxf(v, 0.f);
    }
}

// ---------------------------------------------------------------------------
extern "C" void kernel_launch(void* const* d_in, const int* in_sizes, int n_in,
                              void* d_out, int out_size, void* d_ws, size_t ws_size,
                              hipStream_t stream)
{
    (void)in_sizes; (void)n_in; (void)out_size; (void)ws_size;

    const float* x  = (const float*)d_in[0];
    const int*   ei = (const int*)d_in[1];     // [2, E] row-major: src then dst
    const float* W[4]  = {(const float*)d_in[3],  (const float*)d_in[7],
                          (const float*)d_in[11], (const float*)d_in[15]};
    const float* AS[4] = {(const float*)d_in[4],  (const float*)d_in[8],
                          (const float*)d_in[12], (const float*)d_in[16]};
    const float* AD[4] = {(const float*)d_in[5],  (const float*)d_in[9],
                          (const float*)d_in[13], (const float*)d_in[17]};
    const float* BB[4] = {(const float*)d_in[6],  (const float*)d_in[10],
                          (const float*)d_in[14], (const float*)d_in[18]};
    const float* LW[4] = {(const float*)d_in[19], (const float*)d_in[21],
                          (const float*)d_in[23], (const float*)d_in[25]};
    const float* LB[4] = {(const float*)d_in[20], (const float*)d_in[22],
                          (const float*)d_in[24], (const float*)d_in[26]};

    const int Nn   = 65536;
    const int Ein  = Nn * 16;
    const int Etot = Ein + Nn;           // input edges + self-loops
    const int Kd[4] = {336, 250, 150, 100};
    const int Cd[4] = {125,  75,  50,  30};

    float* w = (float*)d_ws;
    const size_t NB = (size_t)Nn * 250;
    float* hbuf = w;                         // [N, up to 250]
    float* agg  = w + NB;                    // [N, up to 250]
    float* als  = w + 2 * NB;                // [N, 2]
    float* ald  = als + (size_t)Nn * 2;
    float* mm   = ald + (size_t)Nn * 2;
    float* den  = mm  + (size_t)Nn * 2;
    float* esc  = den + (size_t)Nn * 2;      // [E_tot, 2]

    const float* fin = x;
    for (int L = 0; L < 4; ++L) {
        const int K = Kd[L], C = Cd[L], HC = 2 * C;
        const size_t NHC = (size_t)Nn * HC;

        dim3 gg(Nn / TM, (HC + TN - 1) / TN);
        gemm_wmma_f32<<<gg, 128, 0, stream>>>(fin, W[L], nullptr, hbuf, Nn, HC, K, 0);
        gat_scores<<<(Nn + 3) / 4, 128, 0, stream>>>(hbuf, AS[L], AD[L], als, ald, Nn, C);
        gat_reset<<<(unsigned)((NHC + 255) / 256), 256, 0, stream>>>(mm, den, agg, Nn * 2, NHC);
        edge_max_k<<<(Etot + 255) / 256, 256, 0, stream>>>(ei, als, ald, mm, esc, Ein, Etot);
        edge_expsum_k<<<(Etot + 255) / 256, 256, 0, stream>>>(ei, mm, esc, den, Ein, Etot);
        edge_scatter_k<<<(Etot + 3) / 4, 128, 0, stream>>>(ei, esc, den, hbuf, agg, Ein, Etot, C);
        bias_relu_k<<<(unsigned)((NHC + 255) / 256), 256, 0, stream>>>(agg, BB[L], NHC, HC);
        fin = agg;
    }

    // MLP head: agg is [65536,60] == [8192,480] contiguous.
    const int Mm = 8192;
    gemm_wmma_f32<<<dim3(Mm / TM, (200 + TN - 1) / TN), 128, 0, stream>>>(
        agg, LW[0], LB[0], hbuf, Mm, 200, 480, 1);
    gemm_wmma_f32<<<dim3(Mm / TM, (100 + TN - 1) / TN), 128, 0, stream>>>(
        hbuf, LW[1], LB[1], agg, Mm, 100, 200, 1);
    gemm_wmma_f32<<<dim3(Mm / TM, (100 + TN - 1) / TN), 128, 0, stream>>>(
        agg, LW[2], LB[2], hbuf, Mm, 100, 100, 1);
    gemm_wmma_f32<<<dim3(Mm / TM, (29 + TN - 1) / TN), 128, 0, stream>>>(
        hbuf, LW[3], LB[3], (float*)d_out, Mm, 29, 100, 0);
}